// QuanvolutionNet_65481071402721
// MI455X (gfx1250) — compile-verified
//
#include <hip/hip_runtime.h>
#include <math.h>

#define B_SZ   8192
#define NPATCH 196
#define NCLS   10
#define FEAT   784   // NPATCH * 4 wires = GEMM K
#define NPAD   16    // classes padded to WMMA N

typedef __attribute__((ext_vector_type(2))) float v2f;
typedef __attribute__((ext_vector_type(8))) float v8f;

// ---------------------------------------------------------------------------
// Phase 1: quanvolution features. One thread = one 2x2 patch -> 4 <Z_w>.
// 4-qubit statevector (16 fp32 amplitudes) lives entirely in VGPRs.
// ---------------------------------------------------------------------------
__global__ __launch_bounds__(256) void quanv_kernel(
    const float* __restrict__ x,      // (B,1,28,28)
    const float* __restrict__ theta,  // (2,4)
    float* __restrict__ qfeat)        // (B, 784) = (B, NPATCH*4)
{
    int n = blockIdx.x * blockDim.x + threadIdx.x;
    if (n >= B_SZ * NPATCH) return;
    int b = n / NPATCH;
    int p = n - b * NPATCH;
    int r = p / 14, c = p - r * 14;

    const float* xb = x + (size_t)b * 784;
    float px[4];
    px[0] = xb[(2 * r) * 28 + 2 * c];
    px[1] = xb[(2 * r) * 28 + 2 * c + 1];
    px[2] = xb[(2 * r + 1) * 28 + 2 * c];
    px[3] = xb[(2 * r + 1) * 28 + 2 * c + 1];

    // RY(x_w)|0> per wire
    float vc[4], vs[4];
#pragma unroll
    for (int w = 0; w < 4; ++w) {
        float ss, cc;
        sincosf(0.5f * px[w], &ss, &cc);
        vc[w] = cc; vs[w] = ss;
    }

    // product state: idx bits (wire0=bit3 .. wire3=bit0)
    float s[16];
#pragma unroll
    for (int i = 0; i < 16; ++i) {
        float a0 = ((i >> 3) & 1) ? vs[0] : vc[0];
        float a1 = ((i >> 2) & 1) ? vs[1] : vc[1];
        float a2 = ((i >> 1) & 1) ? vs[2] : vc[2];
        float a3 = (i & 1)        ? vs[3] : vc[3];
        s[i] = a0 * a1 * a2 * a3;
    }

    // variational angles (8 values, L2-resident)
    float tc[2][4], ts[2][4];
#pragma unroll
    for (int d = 0; d < 2; ++d)
#pragma unroll
        for (int w = 0; w < 4; ++w) {
            float ss, cc;
            sincosf(0.5f * theta[d * 4 + w], &ss, &cc);
            tc[d][w] = cc; ts[d][w] = ss;
        }

#pragma unroll
    for (int d = 0; d < 2; ++d) {
#pragma unroll
        for (int w = 0; w < 4; ++w) {
            int m = 8 >> w;                  // bit of this wire
            float cR = tc[d][w], sR = ts[d][w];
#pragma unroll
            for (int i = 0; i < 16; ++i) {
                if (i & m) continue;
                int j = i | m;
                float a0 = s[i], a1 = s[j];
                s[i] = cR * a0 - sR * a1;
                s[j] = sR * a0 + cR * a1;
            }
        }
        // CNOT(0->1): control bit3, target bit2
#pragma unroll
        for (int i = 8; i < 12; ++i) { float t = s[i]; s[i] = s[i | 4]; s[i | 4] = t; }
        // CNOT(2->3): control bit1, target bit0
#pragma unroll
        for (int ii = 0; ii < 4; ++ii) {
            int i = (ii << 2) | 2;
            float t = s[i]; s[i] = s[i | 1]; s[i | 1] = t;
        }
    }

    // <Z_w> = sum_i (+/-) |s_i|^2
    float z0 = 0.f, z1 = 0.f, z2 = 0.f, z3 = 0.f;
#pragma unroll
    for (int i = 0; i < 16; ++i) {
        float pr = s[i] * s[i];
        z0 += ((i >> 3) & 1) ? -pr : pr;
        z1 += ((i >> 2) & 1) ? -pr : pr;
        z2 += ((i >> 1) & 1) ? -pr : pr;
        z3 += (i & 1)        ? -pr : pr;
    }
    *(float4*)(qfeat + (size_t)n * 4) = make_float4(z0, z1, z2, z3);
}

// ---------------------------------------------------------------------------
// Pad W (10,784) -> Wp (16,784), rows 10..15 zero. One-shot, 50 KB.
// ---------------------------------------------------------------------------
__global__ __launch_bounds__(256) void padw_kernel(
    const float* __restrict__ W, float* __restrict__ Wp)
{
    int i = blockIdx.x * blockDim.x + threadIdx.x;
    if (i >= NPAD * FEAT) return;
    int n = i / FEAT;
    Wp[i] = (n < NCLS) ? W[i] : 0.0f;
}

// ---------------------------------------------------------------------------
// Phase 2: logits = qfeat @ Wp^T + b, then log_softmax, fp32 WMMA.
// One wave per 32-row tile (two 16x16 C accumulators sharing one B fragment),
// double-buffered loads so WMMAs overlap the next fragment fetch.
// A 16x4 layout: lanes<16 hold (K+0,K+1), lanes>=16 hold (K+2,K+3), M=lane&15.
// B  4x16 layout: same K split per half-wave, N=lane&15.
// C: acc[r] = element (M = r + 8*(lane>=16), N = lane&15).
// ---------------------------------------------------------------------------
__global__ __launch_bounds__(256) void gemm_lsm_kernel(
    const float* __restrict__ qfeat,  // (8192, 784)
    const float* __restrict__ Wp,     // (16, 784), rows >=10 zero
    const float* __restrict__ bias,   // (10,)
    float* __restrict__ out)          // (8192, 10) log-probs
{
    int wave = (blockIdx.x * blockDim.x + threadIdx.x) >> 5;   // tile id, 0..255
    int lane = threadIdx.x & 31;
    int base = wave * 32;          // first output row of this 32-row tile
    int half = lane >> 4;          // 0 -> K+0/1, 1 -> K+2/3
    int l15  = lane & 15;

    const float* A0 = qfeat + (size_t)(base + l15) * FEAT + half * 2;
    const float* A1 = A0 + (size_t)16 * FEAT;
    const float* Br = Wp + (size_t)l15 * FEAT + half * 2;

    v8f acc0 = (v8f)0.0f;
    v8f acc1 = (v8f)0.0f;

    v2f a0 = *(const v2f*)A0;
    v2f a1 = *(const v2f*)A1;
    v2f b  = *(const v2f*)Br;

    for (int k = 4; k < FEAT; k += 4) {
        v2f na0 = *(const v2f*)(A0 + k);
        v2f na1 = *(const v2f*)(A1 + k);
        v2f nb  = *(const v2f*)(Br + k);
        acc0 = __builtin_amdgcn_wmma_f32_16x16x4_f32(
            false, a0, false, b, (short)0, acc0, false, false);
        acc1 = __builtin_amdgcn_wmma_f32_16x16x4_f32(
            false, a1, false, b, (short)0, acc1, false, false);
        a0 = na0; a1 = na1; b = nb;
    }
    acc0 = __builtin_amdgcn_wmma_f32_16x16x4_f32(
        false, a0, false, b, (short)0, acc0, false, false);
    acc1 = __builtin_amdgcn_wmma_f32_16x16x4_f32(
        false, a1, false, b, (short)0, acc1, false, false);

    int  nClmp = (l15 < NCLS) ? l15 : 0;
    float lbias = (l15 < NCLS) ? bias[nClmp] : 0.0f;

#pragma unroll
    for (int t = 0; t < 2; ++t) {
        v8f acc = t ? acc1 : acc0;
        int rbase = base + t * 16;
#pragma unroll
        for (int r = 0; r < 8; ++r) {
            float logit = acc[r] + lbias;
            // row-wise max over the 16 lanes of this half-wave (cols >=10 masked)
            float mval = (l15 < NCLS) ? logit : -__builtin_inff();
#pragma unroll
            for (int off = 1; off < 16; off <<= 1)
                mval = fmaxf(mval, __shfl_xor(mval, off, 32));
            float e = (l15 < NCLS) ? expf(logit - mval) : 0.0f;
            float ssum = e;
#pragma unroll
            for (int off = 1; off < 16; off <<= 1)
                ssum += __shfl_xor(ssum, off, 32);
            float lse = mval + logf(ssum);
            if (l15 < NCLS) {
                int m = r + half * 8;
                out[(size_t)(rbase + m) * NCLS + l15] = logit - lse;
            }
        }
    }
}

extern "C" void kernel_launch(void* const* d_in, const int* in_sizes, int n_in,
                              void* d_out, int out_size, void* d_ws, size_t ws_size,
                              hipStream_t stream) {
    const float* x     = (const float*)d_in[0];  // (8192,1,28,28)
    const float* theta = (const float*)d_in[1];  // (2,4)
    const float* W     = (const float*)d_in[2];  // (10,784)
    const float* bias  = (const float*)d_in[3];  // (10,)
    float* out = (float*)d_out;                  // (8192,10)

    float* qfeat = (float*)d_ws;                      // 8192*784 fp32 = 25.69 MB
    float* Wp    = qfeat + (size_t)B_SZ * FEAT;       // 16*784 fp32  = 50 KB

    int npt = B_SZ * NPATCH;                     // 1,605,632 patches
    quanv_kernel<<<(npt + 255) / 256, 256, 0, stream>>>(x, theta, qfeat);
    padw_kernel<<<(NPAD * FEAT + 255) / 256, 256, 0, stream>>>(W, Wp);

    // 256 waves, each a 32-row tile; 8 waves (256 threads) per block -> 32 blocks
    gemm_lsm_kernel<<<32, 256, 0, stream>>>(qfeat, Wp, bias, out);
}